// OIMLossSafe_9105330667997
// MI455X (gfx1250) — compile-verified
//
#include <hip/hip_runtime.h>
#include <math.h>

// ---------------------------------------------------------------------------
// OIM loss (Online Instance Matching) for MI455X / gfx1250.
//   inputs [1024,256] f32, label [1024] i32 (50000 == ignore),
//   lut [50000,256] f32, cq [5000,256] f32, header i32
// out: [0] loss, [1 .. 12800000] new_lut, [.. +1280000] new_cq
//
// Compute core: fp32 WMMA (v_wmma_f32_16x16x4_f32), fused online-softmax
// partials so the 1024x55000 logits matrix is never materialized.
// ---------------------------------------------------------------------------

#define NFEAT   256
#define NPIDS   50000
#define CQSZ    5000
#define NBATCH  1024
#define NCOLS   55000           // NPIDS + CQSZ
#define NCOLS_P 55040           // padded to CHUNK multiple (pad flag = 2)
#define OIM_S   30.0f
#define CHUNK   64              // columns per workgroup (= per partial chunk)
#define NCHUNK  860             // ceil(55000/64)
#define ROWS_WG 128
#define L2EPS   1e-12f

typedef __attribute__((ext_vector_type(2))) float v2f;
typedef __attribute__((ext_vector_type(8))) float v8f;

// workspace layout (bytes):
//   [0, 1048576)              x_norm [1024][256] f32
//   [1048576, +55040)         column flags u8 (0=ok, 1=bad/all-zero, 2=pad)
//   [1103872, +7045120)       partials float2 [1024][860]  (rowmax, sumexp)
#define WS_XN   0
#define WS_BAD  1048576
#define WS_PART 1103872

__device__ inline float waveReduceSum(float v) {
#pragma unroll
  for (int m = 16; m; m >>= 1) v += __shfl_xor(v, m, 32);
  return v;
}
__device__ inline float halfReduceMax(float v) {   // within each 16-lane half
#pragma unroll
  for (int m = 1; m < 16; m <<= 1) v = fmaxf(v, __shfl_xor(v, m, 32));
  return v;
}
__device__ inline float halfReduceSum(float v) {
#pragma unroll
  for (int m = 1; m < 16; m <<= 1) v += __shfl_xor(v, m, 32);
  return v;
}

// ---------------------------------------------------------------------------
// 1) L2-normalize inputs (wave per row); thread 0 of block 0 zeroes the loss.
// ---------------------------------------------------------------------------
__global__ __launch_bounds__(256) void oim_normalize(
    const float* __restrict__ in, float* __restrict__ xn, float* __restrict__ loss)
{
  if (blockIdx.x == 0 && threadIdx.x == 0) loss[0] = 0.0f;
  const int w = threadIdx.x >> 5, lane = threadIdx.x & 31;
  const int i = blockIdx.x * 8 + w;                 // 128 blocks * 8 waves = 1024 rows
  const float* src = in + (size_t)i * NFEAT;
  float v[8], ss = 0.0f;
#pragma unroll
  for (int u = 0; u < 8; ++u) { v[u] = src[lane + 32 * u]; ss += v[u] * v[u]; }
  ss = waveReduceSum(ss);
  const float inv = 1.0f / fmaxf(sqrtf(ss), L2EPS);
  float* dst = xn + (size_t)i * NFEAT;
#pragma unroll
  for (int u = 0; u < 8; ++u) dst[lane + 32 * u] = v[u] * inv;
}

// ---------------------------------------------------------------------------
// 2) column flags: 1 if row j (lut for j<NPIDS, else cq) is all-zero,
//    2 for the padded tail [55000, 55040) so the GEMM needs no bounds branch.
// ---------------------------------------------------------------------------
__global__ __launch_bounds__(256) void oim_badflags(
    const float* __restrict__ lut, const float* __restrict__ cq,
    unsigned char* __restrict__ flags)
{
  const int w = threadIdx.x >> 5, lane = threadIdx.x & 31;
  const int j = blockIdx.x * 8 + w;                 // 6880 blocks -> 55040 rows
  if (j >= NCOLS) {
    if (lane == 0) flags[j] = 2;                    // pad sentinel
    return;
  }
  const float* src = (j < NPIDS) ? (lut + (size_t)j * NFEAT)
                                 : (cq + (size_t)(j - NPIDS) * NFEAT);
  int nz = 0;
#pragma unroll
  for (int u = 0; u < 8; ++u) nz |= (src[lane + 32 * u] != 0.0f);
  const int any_nz = __any(nz);
  if (lane == 0) flags[j] = (unsigned char)(!any_nz);
}

// ---------------------------------------------------------------------------
// 3) fp32 WMMA GEMM + fused online-softmax partials.
//    Grid (860, 8); block 256 = 8 waves; dynamic LDS 192 KB (A 128KB, B 64KB).
//    Wave w computes rows [rBase+16w, +16) x cols [jBase, +64) (4 WMMA tiles).
//    A 16x4 f32 layout: lane holds row M=lane%16, K pair (k0+2*(lane/16))+{0,1}
//    B 4x16 f32 layout: lane holds col N=lane%16, same K pair (symmetric).
//    C 16x16: VGPR v, lane -> row v+8*(lane/16), col lane%16.
// ---------------------------------------------------------------------------
__global__ __launch_bounds__(256) void oim_gemm_partials(
    const float* __restrict__ xn, const float* __restrict__ lut,
    const float* __restrict__ cq, const unsigned char* __restrict__ flags,
    float2* __restrict__ partials)
{
  extern __shared__ float smem[];
  float* As = smem;                       // [128][256]
  float* Bs = smem + ROWS_WG * NFEAT;     // [64][256]
  const int tid  = threadIdx.x;
  const int w    = tid >> 5, lane = tid & 31;
  const int half = lane >> 4, l16 = lane & 15;
  const int rBase = blockIdx.y * ROWS_WG;
  const int jBase = blockIdx.x * CHUNK;

  // Stage A slab (x_norm rows rBase..rBase+127) -> LDS, 128-bit chunks.
  {
    const float4* src = (const float4*)(xn + (size_t)rBase * NFEAT);
    float4* dst = (float4*)As;
#pragma unroll
    for (int u = 0; u < 32; ++u) dst[tid + 256 * u] = src[tid + 256 * u];
  }
  // Stage B slab (lut/cq rows jBase..jBase+63) -> LDS; zero past 55000.
  {
    float4* dst = (float4*)Bs;
#pragma unroll
    for (int u = 0; u < 16; ++u) {
      const int f  = tid + 256 * u;       // float4 index, < 4096
      const int j  = jBase + (f >> 6);    // 64 float4 per 256-float row
      const int c4 = f & 63;
      float4 v = make_float4(0.f, 0.f, 0.f, 0.f);
      if (j < NPIDS)      v = ((const float4*)(lut + (size_t)j * NFEAT))[c4];
      else if (j < NCOLS) v = ((const float4*)(cq + (size_t)(j - NPIDS) * NFEAT))[c4];
      dst[f] = v;
    }
  }
  __syncthreads();

  v8f acc[4] = {};
  const float* arow = As + (w * 16 + l16) * NFEAT;
  const float* bcol = Bs + l16 * NFEAT;
  for (int k0 = 0; k0 < NFEAT; k0 += 4) {
    const int ks = k0 + 2 * half;
    const v2f a = *(const v2f*)(arow + ks);
#pragma unroll
    for (int t = 0; t < 4; ++t) {
      const v2f b = *(const v2f*)(bcol + t * 16 * NFEAT + ks);
      acc[t] = __builtin_amdgcn_wmma_f32_16x16x4_f32(
          false, a, false, b, (short)0, acc[t], false, false);
    }
  }

  // Branchless column masking (flags padded to 55040): pick the substitute
  // logit per tile, then per-row max & sum(exp(.-max)) over this 64-col chunk.
  float sub[4];   // value to use when flag != 0
  bool  msk[4];   // flag != 0 ?
#pragma unroll
  for (int t = 0; t < 4; ++t) {
    const unsigned char f = flags[jBase + t * 16 + l16];
    msk[t] = (f != 0);
    sub[t] = (f == 2) ? -1e30f : -OIM_S;  // pad -> -inf-ish, bad row -> -30
  }
  float rm[8], rs[8];
#pragma unroll
  for (int v = 0; v < 8; ++v) {
    float mx = -1e30f;
#pragma unroll
    for (int t = 0; t < 4; ++t) {
      const float val = msk[t] ? sub[t] : OIM_S * acc[t][v];
      mx = fmaxf(mx, val);
    }
    rm[v] = halfReduceMax(mx);
    float s = 0.0f;
#pragma unroll
    for (int t = 0; t < 4; ++t) {
      const float val = msk[t] ? sub[t] : OIM_S * acc[t][v];
      s += __expf(val - rm[v]);
    }
    rs[v] = halfReduceSum(s);
  }
  if (l16 == 0) {                          // lanes 0 and 16 write 8 rows each
#pragma unroll
    for (int v = 0; v < 8; ++v) {
      const int row = rBase + w * 16 + v + 8 * half;
      partials[(size_t)row * NCHUNK + blockIdx.x] = make_float2(rm[v], rs[v]);
    }
  }
}

// ---------------------------------------------------------------------------
// 4) Per-row logsumexp merge + target logit + loss accumulation (wave/row).
// ---------------------------------------------------------------------------
__global__ __launch_bounds__(256) void oim_finalize(
    const float* __restrict__ xn, const float* __restrict__ lut,
    const int* __restrict__ label, const unsigned char* __restrict__ flags,
    const float2* __restrict__ partials, float* __restrict__ loss)
{
  const int w = threadIdx.x >> 5, lane = threadIdx.x & 31;
  const int i = blockIdx.x * 8 + w;
  const int y = label[i];
  const bool labeled = (y < NPIDS);
  const int lc = labeled ? y : (NPIDS - 1);
  const bool bad_pos = labeled && (flags[lc] != 0);

  // target logit: 30 * dot(x_i, lut[lc])  (overridden to +30 if bad_pos)
  const float* xr = xn + (size_t)i * NFEAT;
  const float* lr = lut + (size_t)lc * NFEAT;
  float d = 0.0f;
#pragma unroll
  for (int u = 0; u < 8; ++u) d += xr[lane + 32 * u] * lr[lane + 32 * u];
  d = waveReduceSum(d);
  const float tgt = bad_pos ? OIM_S : OIM_S * d;

  // streaming (max, sum) merge over 860 chunks
  float M = -1e30f, S = 0.0f;
  for (int cc = lane; cc < NCHUNK; cc += 32) {
    const float2 p = partials[(size_t)i * NCHUNK + cc];
    const float nM = fmaxf(M, p.x);
    S = S * __expf(M - nM) + p.y * __expf(p.x - nM);
    M = nM;
  }
#pragma unroll
  for (int m = 1; m < 32; m <<= 1) {
    const float oM = __shfl_xor(M, m, 32), oS = __shfl_xor(S, m, 32);
    const float nM = fmaxf(M, oM);
    S = S * __expf(M - nM) + oS * __expf(oM - nM);
    M = nM;
  }
  if (bad_pos) {   // masked -1 at (i,lc) was counted; replace with +1 (x30)
    const float nM = fmaxf(M, OIM_S);
    S = S * __expf(M - nM) + __expf(OIM_S - nM);
    M = nM;
    S -= __expf(-OIM_S - M);
  }
  const float nll = M + logf(S) - tgt;
  const float per = labeled ? nll : 0.0f;
  if (lane == 0) atomicAdd(loss, per * (1.0f / NBATCH));
}

// ---------------------------------------------------------------------------
// 5) copy lut+cq -> output (output is offset by 1 float -> scalar copies).
// ---------------------------------------------------------------------------
__global__ __launch_bounds__(256) void oim_copy(
    const float* __restrict__ lut, const float* __restrict__ cq,
    float* __restrict__ out)   // out = d_out + 1
{
  const size_t idx = (size_t)blockIdx.x * 256 + threadIdx.x;  // 55000 blocks
  const size_t nl = (size_t)NPIDS * NFEAT;
  out[idx] = (idx < nl) ? lut[idx] : cq[idx - nl];
}

// ---------------------------------------------------------------------------
// 6) Sequential-equivalent EMA of lut rows. The wave owning the FIRST batch
//    occurrence of a label walks its whole chain in batch order (distinct
//    labels are independent, so this matches the reference scan exactly).
// ---------------------------------------------------------------------------
__global__ __launch_bounds__(256) void oim_lut_ema(
    const float* __restrict__ xn, const int* __restrict__ label,
    const float* __restrict__ lut, float* __restrict__ out_lut)
{
  const int w = threadIdx.x >> 5, lane = threadIdx.x & 31;
  const int i = blockIdx.x * 8 + w;
  const int y = label[i];
  if (y >= NPIDS) return;
  for (int j = 0; j < i; ++j)
    if (label[j] == y) return;            // not the first occurrence
  float r[8];
  const float* src = lut + (size_t)y * NFEAT;
#pragma unroll
  for (int u = 0; u < 8; ++u) r[u] = src[lane + 32 * u];
  for (int j = i; j < NBATCH; ++j) {
    if (label[j] != y) continue;
    const float* xr = xn + (size_t)j * NFEAT;
    float ss = 0.0f;
#pragma unroll
    for (int u = 0; u < 8; ++u) {
      r[u] = 0.5f * r[u] + 0.5f * xr[lane + 32 * u];
      ss += r[u] * r[u];
    }
    ss = waveReduceSum(ss);
    const float inv = 1.0f / fmaxf(sqrtf(ss), L2EPS);
#pragma unroll
    for (int u = 0; u < 8; ++u) r[u] *= inv;
  }
  float* dst = out_lut + (size_t)y * NFEAT;
#pragma unroll
  for (int u = 0; u < 8; ++u) dst[lane + 32 * u] = r[u];
}

// ---------------------------------------------------------------------------
// 7) cq circular-queue writes: slot = (header + rank-among-unlabeled) % CQSZ.
//    #unlabeled <= 1024 < CQSZ, so slots are unique -> fully parallel.
// ---------------------------------------------------------------------------
__global__ __launch_bounds__(256) void oim_cq_write(
    const float* __restrict__ xn, const int* __restrict__ label,
    const int* __restrict__ header, float* __restrict__ out_cq)
{
  const int w = threadIdx.x >> 5, lane = threadIdx.x & 31;
  const int i = blockIdx.x * 8 + w;
  if (label[i] < NPIDS) return;
  int rank = 0;
  for (int j = 0; j < i; ++j) rank += (label[j] >= NPIDS);
  const int slot = (header[0] + rank) % CQSZ;
  const float* xr = xn + (size_t)i * NFEAT;
  float* dst = out_cq + (size_t)slot * NFEAT;
#pragma unroll
  for (int u = 0; u < 8; ++u) dst[lane + 32 * u] = xr[lane + 32 * u];
}

// ---------------------------------------------------------------------------
extern "C" void kernel_launch(void* const* d_in, const int* in_sizes, int n_in,
                              void* d_out, int out_size, void* d_ws, size_t ws_size,
                              hipStream_t stream) {
  (void)in_sizes; (void)n_in; (void)out_size; (void)ws_size;
  const float* inputs = (const float*)d_in[0];
  const int*   label  = (const int*)d_in[1];
  const float* lut    = (const float*)d_in[2];
  const float* cq     = (const float*)d_in[3];
  const int*   header = (const int*)d_in[4];
  float* out = (float*)d_out;              // [0]=loss, +1 lut, +1+12.8M cq

  char* ws = (char*)d_ws;                  // needs ~8.2 MB
  float*         xn       = (float*)(ws + WS_XN);
  unsigned char* flags    = (unsigned char*)(ws + WS_BAD);
  float2*        partials = (float2*)(ws + WS_PART);

  const int gemm_lds = (ROWS_WG + CHUNK) * NFEAT * (int)sizeof(float); // 196608 B
  (void)hipFuncSetAttribute((const void*)oim_gemm_partials,
                            hipFuncAttributeMaxDynamicSharedMemorySize, gemm_lds);

  oim_normalize<<<128, 256, 0, stream>>>(inputs, xn, out);
  oim_badflags<<<NCOLS_P / 8, 256, 0, stream>>>(lut, cq, flags);
  oim_gemm_partials<<<dim3(NCHUNK, NBATCH / ROWS_WG), 256, gemm_lds, stream>>>(
      xn, lut, cq, flags, partials);
  oim_finalize<<<128, 256, 0, stream>>>(xn, lut, label, flags, partials, out);
  oim_copy<<<NCOLS, 256, 0, stream>>>(lut, cq, out + 1);
  oim_lut_ema<<<128, 256, 0, stream>>>(xn, label, lut, out + 1);
  oim_cq_write<<<128, 256, 0, stream>>>(xn, label, header,
                                        out + 1 + (size_t)NPIDS * NFEAT);
}